// StandardMultiHeadTriadNodeNet_59734405153440
// MI455X (gfx1250) — compile-verified
//
#include <hip/hip_runtime.h>
#include <math.h>

// ---- problem constants (from reference: B=2, N=1024, D=64, H=16, HD=4, NIN=12) ----
#define B_   2
#define N_   1024
#define D_   64
#define H_   16
#define HD_  4
#define NIN_ 12
#define NN_  (N_ * N_)      // 1048576 per (b,h) slice
#define BH_  (B_ * H_)      // 32

typedef __attribute__((ext_vector_type(2))) float        v2f;
typedef __attribute__((ext_vector_type(8))) float        v8f;
typedef __attribute__((ext_vector_type(4))) unsigned int v4u;
typedef __attribute__((ext_vector_type(8))) int          v8i;
typedef __attribute__((ext_vector_type(4))) int          v4i;

#if defined(__has_builtin)
#  if __has_builtin(__builtin_amdgcn_tensor_load_to_lds)
#    define HAVE_TDM 1
#  endif
#endif
#ifndef HAVE_TDM
#  define HAVE_TDM 0
#endif

__device__ __forceinline__ float buntanh_f(float x) {
    const float U = 0.1f;
    const float L = 1.8477590650225735f;  // sqrt(2 + sqrt(2))
    return (1.0f - U) * L * tanhf(x / L) + U * x;
}

// ---------------------------------------------------------------------------
// k_zero: clear the tiny accumulator block (rms sumsq / var sum / var sumsq)
// and the loss slot in d_out. Runs every launch -> deterministic.
// ---------------------------------------------------------------------------
__global__ void k_zero(float* __restrict__ accs, float* __restrict__ loss_slot) {
    int t = threadIdx.x;
    if (t < 96) accs[t] = 0.0f;
    if (t == 0) *loss_slot = 0.0f;
}

// ---------------------------------------------------------------------------
// k_node_fwd: per-node GEMV x(1x64) * W(64x64) for w1/w2/w3 + buntanh.
// 4 nodes per 256-thread block; thread = output element; W rows stream
// coalesced (64 consecutive floats per row). HBM-bound: 100 MB of weights.
// ---------------------------------------------------------------------------
__global__ void k_node_fwd(const float* __restrict__ state,
                           const float* __restrict__ w1,
                           const float* __restrict__ w2,
                           const float* __restrict__ w3,
                           float* __restrict__ pred,
                           float* __restrict__ keys,
                           float* __restrict__ qrs) {
    __shared__ float xs[4][D_];
    int tid = threadIdx.x;
    int q = tid >> 6, e = tid & 63;
    int g = blockIdx.x * 4 + q;               // global node id in [0, B*N)
    size_t gbase = (size_t)g * D_;
    xs[q][e] = state[gbase + e];
    __syncthreads();

    size_t wbase = (size_t)g * D_ * D_;
    const float* W1 = w1 + wbase;
    const float* W2 = w2 + wbase;
    const float* W3 = w3 + wbase;
    // speculative prefetch of the next node's weights (gfx1250 global_prefetch_b8)
    __builtin_prefetch(W1 + D_ * D_, 0, 1);
    __builtin_prefetch(W2 + D_ * D_, 0, 1);
    __builtin_prefetch(W3 + D_ * D_, 0, 1);

    const float* xr = xs[q];
    float a1 = 0.0f, a2 = 0.0f, a3 = 0.0f;
#pragma unroll 8
    for (int d = 0; d < D_; ++d) {
        float xv = xr[d];
        a1 = fmaf(xv, W1[d * D_ + e], a1);
        a2 = fmaf(xv, W2[d * D_ + e], a2);
        a3 = fmaf(xv, W3[d * D_ + e], a3);
    }
    pred[gbase + e] = buntanh_f(a1);   // w1 -> prediction
    keys[gbase + e] = buntanh_f(a2);   // w2 -> keys
    qrs [gbase + e] = buntanh_f(a3);   // w3 -> queries
}

// ---------------------------------------------------------------------------
// k_qk_wmma: raw_A[b,h] = Q K^T / sqrt(HD) via V_WMMA_F32_16X16X4_F32.
// K-dim == HD == 4: exact fit, one WMMA per 16x16 tile, f32 end-to-end.
// Also accumulates sum(raw_A^2) per (b,h) for rms_norm.
// A operand (16x4 f32): lane m=l&15; lanes<16 hold K=0,1; lanes>=16 hold K=2,3.
// B operand (4x16): lane col j=l&15; lanes<16 rows 0,1; lanes>=16 rows 2,3.
// ---------------------------------------------------------------------------
__global__ void k_qk_wmma(const float* __restrict__ qrs,
                          const float* __restrict__ keys,
                          float* __restrict__ Amat,
                          float* __restrict__ rms_acc) {
    int lane = threadIdx.x & 31;
    int t = blockIdx.x * 8 + (threadIdx.x >> 5);  // tile id
    int jt = t & 63; t >>= 6;
    int it = t & 63; t >>= 6;
    int h  = t & 15;
    int b  = t >> 4;

    int m    = lane & 15;
    int half = lane >> 4;            // 0 or 1
    int k0   = half * 2;             // K base for this half-wave

    const float* qp = qrs  + ((size_t)(b * N_ + it * 16 + m) * D_ + h * HD_ + k0);
    const float* kp = keys + ((size_t)(b * N_ + jt * 16 + m) * D_ + h * HD_ + k0);
    v2f a, bb;
    a[0]  = qp[0]; a[1]  = qp[1];
    bb[0] = kp[0]; bb[1] = kp[1];

    v8f c = {};
    c = __builtin_amdgcn_wmma_f32_16x16x4_f32(false, a, false, bb, (short)0, c, false, false);

    size_t slice = ((size_t)(b * H_ + h)) * NN_;
    float sq = 0.0f;
#pragma unroll
    for (int r = 0; r < 8; ++r) {
        int mo = r + half * 8;
        float v = c[r] * 0.5f;       // 1/sqrt(HD)
        Amat[slice + (size_t)(it * 16 + mo) * N_ + jt * 16 + m] = v;
        sq += v * v;
    }
    for (int off = 16; off > 0; off >>= 1) sq += __shfl_down(sq, off, 32);
    if (lane == 0) atomicAdd(&rms_acc[b * H_ + h], sq);
}

// ---------------------------------------------------------------------------
// k_ema_rownorm: fused [rms_norm scale + EMA(+row-NIN adj) + mask + relu]
// with sinkhorn-1 row-normalization. One block per (bh,row); the 1024 row
// elements live in 4 regs/thread -> single read + single write of A.
// ---------------------------------------------------------------------------
__global__ void k_ema_rownorm(float* __restrict__ Amat,
                              const float* __restrict__ ema_in,
                              const float* __restrict__ mask_in,
                              const float* __restrict__ rms_acc) {
    __shared__ float red[256];
    int tid = threadIdx.x;
    int row = blockIdx.x;            // bh*N + i
    int bh  = row >> 10;
    int i   = row & (N_ - 1);
    size_t base = (size_t)row * N_;

    float scale   = rsqrtf(rms_acc[bh] * (1.0f / NN_) + 1e-8f);
    float rowmask = (i < NIN_) ? 0.0f : 1.0f;
    bool  adjrow  = (i == NIN_);

    float v[4]; float s = 0.0f;
#pragma unroll
    for (int k = 0; k < 4; ++k) {
        int j = tid + 256 * k;
        float adj   = (adjrow && j < NIN_) ? (3.0f / NIN_) : 0.0f;
        float maskv = rowmask * mask_in[base + j];
        float val = ((ema_in[base + j] + adj) * 0.8f + Amat[base + j] * scale * 0.2f) * maskv;
        val  = fmaxf(val, 0.0f);     // sinkhorn's max(A, 0)
        v[k] = val; s += val;
    }
    red[tid] = s; __syncthreads();
    for (int o = 128; o > 0; o >>= 1) { if (tid < o) red[tid] += red[tid + o]; __syncthreads(); }
    float inv = 1.0f / (red[0] + 1e-8f);
#pragma unroll
    for (int k = 0; k < 4; ++k) Amat[base + tid + 256 * k] = v[k] * inv;
}

// ---------------------------------------------------------------------------
// k_colsum: column sums per (b,h). Block = (bh, 256-column chunk); each
// thread owns a column -> fully coalesced row sweeps.
// ---------------------------------------------------------------------------
__global__ void k_colsum(const float* __restrict__ Amat, float* __restrict__ colsum) {
    int bh = blockIdx.x >> 2;
    int j  = ((blockIdx.x & 3) << 8) + threadIdx.x;
    const float* base = Amat + (size_t)bh * NN_ + j;
    float s = 0.0f;
    for (int i = 0; i < N_; ++i) s += base[(size_t)i * N_];
    colsum[bh * N_ + j] = s;
}

// ---------------------------------------------------------------------------
// k_cs_rownorm: fused column-scale (iter k) + optional sparsify + row-norm
// (iter k+1). Row-resident in registers: one read + one write of A.
// ---------------------------------------------------------------------------
__global__ void k_cs_rownorm(float* __restrict__ Amat,
                             const float* __restrict__ colsum,
                             int sparsify) {
    __shared__ float red[256];
    int tid = threadIdx.x;
    int row = blockIdx.x;
    int bh  = row >> 10;
    size_t base = (size_t)row * N_;
    const float* cs = colsum + bh * N_;
    const float ub = 1.0f / (N_ + 1.0f);

    float v[4]; float s = 0.0f;
#pragma unroll
    for (int k = 0; k < 4; ++k) {
        int j = tid + 256 * k;
        float val = Amat[base + j] / (cs[j] + 1e-8f);
        if (sparsify) val = (val > ub) ? val : 0.0f;
        v[k] = val; s += val;
    }
    red[tid] = s; __syncthreads();
    for (int o = 128; o > 0; o >>= 1) { if (tid < o) red[tid] += red[tid + o]; __syncthreads(); }
    float inv = 1.0f / (red[0] + 1e-8f);
#pragma unroll
    for (int k = 0; k < 4; ++k) Amat[base + tid + 256 * k] = v[k] * inv;
}

// ---------------------------------------------------------------------------
// k_cs_maskvar: last column-scale of sinkhorn-2 + A_mask + variance stats
// (sum, sumsq per (b,h)) for the boredom-noise term vd.
// ---------------------------------------------------------------------------
__global__ void k_cs_maskvar(float* __restrict__ Amat,
                             const float* __restrict__ colsum,
                             const float* __restrict__ mask_in,
                             float* __restrict__ var_sum,
                             float* __restrict__ var_sq) {
    __shared__ float r1[256], r2[256];
    int tid = threadIdx.x;
    int row = blockIdx.x;
    int bh  = row >> 10;
    int i   = row & (N_ - 1);
    size_t base = (size_t)row * N_;
    const float* cs = colsum + bh * N_;
    float rowmask = (i < NIN_) ? 0.0f : 1.0f;

    float s = 0.0f, s2 = 0.0f;
#pragma unroll
    for (int k = 0; k < 4; ++k) {
        int j = tid + 256 * k;
        float val = Amat[base + j] / (cs[j] + 1e-8f) * rowmask * mask_in[base + j];
        Amat[base + j] = val;
        s += val; s2 += val * val;
    }
    r1[tid] = s; r2[tid] = s2; __syncthreads();
    for (int o = 128; o > 0; o >>= 1) {
        if (tid < o) { r1[tid] += r1[tid + o]; r2[tid] += r2[tid + o]; }
        __syncthreads();
    }
    if (tid == 0) { atomicAdd(&var_sum[bh], r1[0]); atomicAdd(&var_sq[bh], r2[0]); }
}

// ---------------------------------------------------------------------------
// k_av_wmma: target = A @ (V + vd*noise) per (b,h), K-chained f32 WMMA.
// All 8 waves of a block share one (b,h): the 16 KB V head-slice (strided,
// stride 64 floats) and the 16 KB contiguous noise slice are staged into LDS
// ONCE per block by the Tensor Data Mover (2D D#, tile 4x1024), completed
// with s_wait_tensorcnt, then vd*noise is folded in-LDS. The 256-iteration
// WMMA loop then feeds B operands from ds_load instead of redundant global
// loads. Only 4 of 16 output columns are live (HD=4) -> masked via value
// selects so EXEC stays all-ones for every WMMA. Softsign fused into store.
// NOTE: `sm` is this kernel's only static LDS allocation, so its group-
// segment offset is 0 — the D#'s lds_addr fields are hardcoded accordingly.
// ---------------------------------------------------------------------------
__global__ void k_av_wmma(const float* __restrict__ Amat,
                          const float* __restrict__ outbuf,
                          const float* __restrict__ noise,
                          const float* __restrict__ var_sum,
                          const float* __restrict__ var_sq,
                          float* __restrict__ tbuf) {
    __shared__ struct { float V[N_ * HD_]; float Nz[N_ * HD_]; } sm;  // 32 KB

    int tid  = threadIdx.x;
    int lane = tid & 31;
    int bh   = blockIdx.x >> 3;                  // uniform per block
    int it   = ((blockIdx.x & 7) << 3) + (tid >> 5);
    int h    = bh & 15;
    int b    = bh >> 4;

    // ---- stage V (strided head slice of output_buf) and noise into LDS ----
#if HAVE_TDM
    if (tid < 32) {                              // one wave issues the TDM ops
        unsigned long long gaV =
            (unsigned long long)(uintptr_t)(outbuf + ((size_t)b * N_) * D_ + h * HD_);
        unsigned long long gaN =
            (unsigned long long)(uintptr_t)(noise + (size_t)bh * N_ * HD_);
        // D# group0: count=1 | lds_addr | global_addr[56:0] | type=2
        v4u g0V = { 1u, 0u,
                    (unsigned int)gaV, (unsigned int)(gaV >> 32) | 0x80000000u };
        v4u g0N = { 1u, (unsigned int)(N_ * HD_ * sizeof(float)),
                    (unsigned int)gaN, (unsigned int)(gaN >> 32) | 0x80000000u };
        // D# group1: data_size=4B; tensor_dim0=4, tensor_dim1=1024;
        //            tile_dim0=4, tile_dim1=1024; dim0 stride 64 (V) / 4 (noise)
        v8i g1V = { 0x00020000, (4 << 16), (1024 << 16), (4 << 16), 1024, 64, 0, 0 };
        v8i g1N = { 0x00020000, (4 << 16), (1024 << 16), (4 << 16), 1024,  4, 0, 0 };
        v4i gz4 = { 0, 0, 0, 0 };                // 2D tile: groups 2/3 unused
        v8i gz8 = { 0, 0, 0, 0, 0, 0, 0, 0 };    // extra descriptor words (unused)
        __builtin_amdgcn_tensor_load_to_lds(g0V, g1V, gz4, gz4, gz8, 0);
        __builtin_amdgcn_tensor_load_to_lds(g0N, g1N, gz4, gz4, gz8, 0);
        __builtin_amdgcn_s_wait_tensorcnt(0);
    }
    __syncthreads();
#else
    for (int k = 0; k < 16; ++k) {
        int i = tid + 256 * k;                   // i = node*4 + jj
        int node = i >> 2, jj2 = i & 3;
        sm.V[i]  = outbuf[((size_t)b * N_ + node) * D_ + h * HD_ + jj2];
        sm.Nz[i] = noise[(size_t)bh * N_ * HD_ + i];
    }
    __syncthreads();
#endif

    // ---- vd = relu(A_VARIANCE - var(A)) ; fold boredom noise into V in LDS ----
    float mean = var_sum[bh] * (1.0f / NN_);
    float var  = var_sq[bh] * (1.0f / NN_) - mean * mean;
    float vd   = fmaxf(0.0026f - var, 0.0f);
#pragma unroll 4
    for (int k = 0; k < 16; ++k) {
        int i = tid + 256 * k;
        sm.V[i] = sm.V[i] + vd * sm.Nz[i];
    }
    __syncthreads();

    // ---- K-chained WMMA: 256 x V_WMMA_F32_16X16X4_F32 over K = 1024 ----
    int m    = lane & 15;
    int half = lane >> 4;
    int j    = lane & 15;
    float jmask = (j < HD_) ? 1.0f : 0.0f;       // only 4 live output columns
    int jj = j & 3;

    const float* arow = Amat + ((size_t)bh * N_ + it * 16 + m) * N_;
    v8f c = {};
    for (int kk = 0; kk < N_ / 4; ++kk) {
        int k0 = kk * 4 + half * 2;              // two K rows per lane
        v2f a, bb;
        a[0] = arow[k0];
        a[1] = arow[k0 + 1];
        bb[0] = jmask * sm.V[k0 * HD_ + jj];
        bb[1] = jmask * sm.V[(k0 + 1) * HD_ + jj];
        c = __builtin_amdgcn_wmma_f32_16x16x4_f32(false, a, false, bb, (short)0, c, false, false);
    }
    if (j < HD_) {
#pragma unroll
        for (int r = 0; r < 8; ++r) {
            int mo = r + half * 8;
            float t = c[r];
            t = t / (1.0f + fabsf(t));           // softsign
            tbuf[((size_t)b * N_ + it * 16 + mo) * D_ + h * HD_ + j] = t;
        }
    }
}

// ---------------------------------------------------------------------------
// k_final: env-row splice + triad losses + prediction output + loss reduce.
// d_out = [prediction (B*N*D floats), total_loss (1 float)]
// ---------------------------------------------------------------------------
__global__ void k_final(const float* __restrict__ env,
                        const float* __restrict__ tbuf,
                        const float* __restrict__ pred,
                        const float* __restrict__ keys,
                        const float* __restrict__ qrs,
                        float* __restrict__ out) {
    __shared__ float red[256];
    int tid = threadIdx.x;
    size_t idx = (size_t)blockIdx.x * 256 + tid;
    int nd = (int)(idx & (N_ * D_ - 1));
    int b  = (int)(idx >> 16);                   // N*D == 65536
    int n  = nd >> 6;
    int d  = nd & 63;

    float tfin = (n < NIN_) ? env[b * (NIN_ * D_) + n * D_ + d] : tbuf[idx];
    float p = pred[idx], k = keys[idx], q = qrs[idx];
    float err1 = p - tfin;
    float lw2  = (k - err1) * (k - err1);
    float err2 = k - err1;
    float lw3  = (q - err2) * (q - err2);
    float err3 = q - err2;
    float d1   = p - (tfin + err3);
    float lw1  = d1 * d1;

    out[idx] = p;                                // prediction (stop_gradient = identity)
    red[tid] = lw1 + lw2 + lw3;
    __syncthreads();
    for (int o = 128; o > 0; o >>= 1) { if (tid < o) red[tid] += red[tid + o]; __syncthreads(); }
    if (tid == 0) atomicAdd(&out[(size_t)B_ * N_ * D_], red[0]);
}

// ---------------------------------------------------------------------------
extern "C" void kernel_launch(void* const* d_in, const int* in_sizes, int n_in,
                              void* d_out, int out_size, void* d_ws, size_t ws_size,
                              hipStream_t stream) {
    (void)in_sizes; (void)n_in; (void)out_size; (void)ws_size;
    const float* env    = (const float*)d_in[0];  // (B, NIN*D)
    const float* state  = (const float*)d_in[1];  // (B, N, D)
    const float* outbuf = (const float*)d_in[2];  // (B, N, D)
    const float* w1     = (const float*)d_in[3];  // (B, N, D, D)
    const float* w2     = (const float*)d_in[4];
    const float* w3     = (const float*)d_in[5];
    const float* ema    = (const float*)d_in[6];  // (B, H, N, N)
    const float* maskI  = (const float*)d_in[7];  // (B, H, N, N)
    const float* noise  = (const float*)d_in[8];  // (B, H, N, HD)
    float* out = (float*)d_out;
    float* ws  = (float*)d_ws;

    // workspace layout (floats); total ~34.1M floats ~= 136.5 MB
    float* Amat    = ws;                               // B*H*N*N
    float* qbuf    = Amat + (size_t)BH_ * NN_;         // B*N*D
    float* kbuf    = qbuf + (size_t)B_ * N_ * D_;
    float* pbuf    = kbuf + (size_t)B_ * N_ * D_;
    float* tbuf    = pbuf + (size_t)B_ * N_ * D_;
    float* colsum  = tbuf + (size_t)B_ * N_ * D_;      // BH*N
    float* accs    = colsum + (size_t)BH_ * N_;        // 96: [rms 32][vsum 32][vsq 32]
    float* rms_acc = accs;
    float* var_sum = accs + 32;
    float* var_sq  = accs + 64;

    k_zero<<<1, 128, 0, stream>>>(accs, out + (size_t)B_ * N_ * D_);

    k_node_fwd<<<B_ * N_ / 4, 256, 0, stream>>>(state, w1, w2, w3, pbuf, kbuf, qbuf);

    k_qk_wmma<<<(B_ * H_ * 64 * 64) / 8, 256, 0, stream>>>(qbuf, kbuf, Amat, rms_acc);

    // sinkhorn #1: row1 fused with ema/mask; then 4x (col, row); final col fused below
    k_ema_rownorm<<<BH_ * N_, 256, 0, stream>>>(Amat, ema, maskI, rms_acc);
    for (int t = 0; t < 4; ++t) {
        k_colsum<<<BH_ * 4, 256, 0, stream>>>(Amat, colsum);
        k_cs_rownorm<<<BH_ * N_, 256, 0, stream>>>(Amat, colsum, 0);
    }
    // sk1 col5 + sparsify + sk2 row1, fused
    k_colsum<<<BH_ * 4, 256, 0, stream>>>(Amat, colsum);
    k_cs_rownorm<<<BH_ * N_, 256, 0, stream>>>(Amat, colsum, 1);
    // sinkhorn #2 middle iterations
    for (int t = 0; t < 4; ++t) {
        k_colsum<<<BH_ * 4, 256, 0, stream>>>(Amat, colsum);
        k_cs_rownorm<<<BH_ * N_, 256, 0, stream>>>(Amat, colsum, 0);
    }
    // sk2 col5 + mask + variance stats, fused
    k_colsum<<<BH_ * 4, 256, 0, stream>>>(Amat, colsum);
    k_cs_maskvar<<<BH_ * N_, 256, 0, stream>>>(Amat, colsum, maskI, var_sum, var_sq);

    k_av_wmma<<<(B_ * H_ * 64) / 8, 256, 0, stream>>>(Amat, outbuf, noise, var_sum, var_sq, tbuf);

    k_final<<<(B_ * N_ * D_) / 256, 256, 0, stream>>>(env, tbuf, pbuf, kbuf, qbuf, out);
}